// SpikeEncoder_45878840656605
// MI455X (gfx1250) — compile-verified
//
#include <hip/hip_runtime.h>

typedef __attribute__((ext_vector_type(2))) float v2f;
typedef __attribute__((ext_vector_type(4))) float f4;
typedef __attribute__((ext_vector_type(8))) float v8f;

#define B_DIM 1024
#define D_DIM 1024
#define N_DIM 1024
#define T_STEPS 128
#define BETA 0.9f
#define THRESH 1.0f

// One wave32 computes one 16x16 tile of h = input @ W^T + b via
// V_WMMA_F32_16X16X4_F32, then runs the 128-step LIF recurrence in
// registers and streams packed spike float4s with non-temporal stores.
__global__ __launch_bounds__(256) void spike_encoder_fused(
    const float* __restrict__ in,    // [B, D]
    const float* __restrict__ W,     // [N, D]
    const float* __restrict__ bias,  // [N]
    float* __restrict__ out)         // [B, N, T]
{
  const int wave = (blockIdx.x << 3) | (threadIdx.x >> 5);  // 0..4095
  const int lane = threadIdx.x & 31;
  const int bt   = wave >> 6;   // M-tile 0..63
  const int nt   = wave & 63;   // N-tile 0..63
  const int half = lane >> 4;   // 0: K=0..1 / M=r ; 1: K=2..3 / M=r+8
  const int lm   = lane & 15;

  // A-frag: lane l loads input[bt*16 + (l&15)][k + 2*(l>>4) .. +1]  (b64)
  // B-frag: lane l loads W    [nt*16 + (l&15)][k + 2*(l>>4) .. +1]  (b64)
  const float* arow = in + (size_t)(bt * 16 + lm) * D_DIM + half * 2;
  const float* brow = W  + (size_t)(nt * 16 + lm) * D_DIM + half * 2;

  v8f c = {};
#pragma unroll 8
  for (int k = 0; k < D_DIM; k += 4) {
    v2f a = *(const v2f*)(arow + k);
    v2f b = *(const v2f*)(brow + k);
    // D = A(16x4,f32) x B(4x16,f32) + C(16x16,f32)
    c = __builtin_amdgcn_wmma_f32_16x16x4_f32(
        /*neg_a=*/false, a, /*neg_b=*/false, b,
        /*c_mod=*/(short)0, c, /*reuse_a=*/false, /*reuse_b=*/false);
  }

  // C/D layout: VGPR r -> M = r + 8*half, N = lm (all 8 regs share N).
  const float bb = bias[nt * 16 + lm];
  float h[8], mem[8];
  float* optr[8];
#pragma unroll
  for (int r = 0; r < 8; ++r) {
    h[r]   = c[r] + bb;
    mem[r] = 0.0f;
    const int row = bt * 16 + half * 8 + r;   // batch index
    const int col = nt * 16 + lm;             // neuron index
    optr[r] = out + ((size_t)row * N_DIM + col) * T_STEPS;
  }

  // Leaky integrate-and-fire, reset_mechanism='subtract':
  //   reset   = (mem_prev > 1)
  //   mem     = beta*mem_prev + h - reset
  //   spike_t = (mem > 1)
  // Sequential in t; 8 elements per lane in parallel; 4 steps packed per store.
  for (int t4 = 0; t4 < T_STEPS / 4; ++t4) {
    f4 sb[8];
#pragma unroll
    for (int j = 0; j < 4; ++j) {
#pragma unroll
      for (int r = 0; r < 8; ++r) {
        const float reset = mem[r] > THRESH ? THRESH : 0.0f;
        mem[r] = fmaf(BETA, mem[r], h[r]) - reset;
        sb[r][j] = mem[r] > THRESH ? 1.0f : 0.0f;
      }
    }
#pragma unroll
    for (int r = 0; r < 8; ++r) {
      __builtin_nontemporal_store(sb[r], (f4*)(optr[r] + t4 * 4));
    }
  }
}

extern "C" void kernel_launch(void* const* d_in, const int* in_sizes, int n_in,
                              void* d_out, int out_size, void* d_ws, size_t ws_size,
                              hipStream_t stream) {
  const float* in   = (const float*)d_in[0];  // [1024, 1024]
  const float* W    = (const float*)d_in[1];  // [1024, 1024]
  const float* bias = (const float*)d_in[2];  // [1024]
  float* out        = (float*)d_out;          // [1024, 1024, 128]

  // 4096 tiles of 16x16, 8 waves (256 threads) per block -> 512 blocks.
  spike_encoder_fused<<<dim3(512), dim3(256), 0, stream>>>(in, W, bias, out);
}